// Policy_48799418417266
// MI455X (gfx1250) — compile-verified
//
#include <hip/hip_runtime.h>

// fp32 WMMA tile types (wave32): A/B = 16x4 / 4x16 (2 VGPRs), C/D = 16x16 (8 VGPRs)
typedef __attribute__((ext_vector_type(2))) float v2f;
typedef __attribute__((ext_vector_type(8))) float v8f;

#define NEG_SLOPE   0.01f
// 256 * 0.5*log(2*pi)
#define LOGP_CONST  235.2482645f

__device__ __forceinline__ v8f wmma4(v2f a, v2f b, v8f c) {
    // D = A(16x4) * B(4x16) + C, fp32 everywhere.
    // 8-arg pattern: (neg_a, A, neg_b, B, c_mod, C, reuse_a, reuse_b)
    return __builtin_amdgcn_wmma_f32_16x16x4_f32(false, a, false, b, (short)0, c, false, false);
}

__device__ __forceinline__ float leaky(float x) {
    return fmaxf(x, 0.0f) + NEG_SLOPE * fminf(x, 0.0f);
}

// One block = 8 waves; each wave computes a 16-row tile of the MLP end-to-end.
__global__ __launch_bounds__(256, 1)
void policy_mlp_kernel(const float* __restrict__ state,   // [B,512]
                       const float* __restrict__ W1,      // [512,64]
                       const float* __restrict__ b1,      // [64]
                       const float* __restrict__ W2,      // [64,32]
                       const float* __restrict__ b2,      // [32]
                       const float* __restrict__ W3,      // [32,256]
                       const float* __restrict__ b3,      // [256]
                       const float* __restrict__ noise,   // [B,256]
                       float* __restrict__ out_act,       // [B,256]
                       float* __restrict__ out_logp,      // [B]
                       int B)
{
    // LDS: small reused weights + per-wave h1/h2 staging (D-layout -> A-layout hop).
    __shared__ __align__(16) float sW2[64 * 32];          //  8 KB
    __shared__ __align__(16) float sb1[64];
    __shared__ __align__(16) float sb2[32];
    __shared__ __align__(16) float sb3[256];
    __shared__ __align__(16) float sh1[8][16 * 66];       // stride 66: even (8B-aligned v2f) + bank-spread
    __shared__ __align__(16) float sh2[8][16 * 34];

    const int tid = threadIdx.x;
    for (int i = tid; i < 64 * 32; i += 256) sW2[i] = W2[i];
    if (tid < 64) sb1[tid] = b1[tid];
    if (tid < 32) sb2[tid] = b2[tid];
    sb3[tid] = b3[tid];                                   // blockDim.x == 256
    __syncthreads();

    const int lane = tid & 31;
    const int wave = tid >> 5;
    const int half = lane >> 4;       // 0: lanes 0-15, 1: lanes 16-31
    const int ln   = lane & 15;       // row (A) / col (B,D) index within tile
    const int koff = half * 2;        // K striping for A/B layout
    const int r0   = (blockIdx.x * 8 + wave) * 16;
    if (r0 >= B) return;

    float* h1w = &sh1[wave][0];
    float* h2w = &sh2[wave][0];

    // ---------------- GEMM1: h1 = leaky(state[16,512] @ W1[512,64] + b1) ----------------
    v8f acc1[4] = {};                 // 4 N-tiles of 16
    const float* srow = state + (size_t)(r0 + ln) * 512;
    for (int k = 0; k < 512; k += 4) {
        v2f a = *(const v2f*)(srow + k + koff);           // A: rows streamed per lane
        #pragma unroll
        for (int nt = 0; nt < 4; ++nt) {
            v2f b;
            b.x = W1[(k + koff)     * 64 + nt * 16 + ln]; // B from global (hot in WGP$/L2)
            b.y = W1[(k + koff + 1) * 64 + nt * 16 + ln];
            acc1[nt] = wmma4(a, b, acc1[nt]);
        }
    }
    // bias + leaky, stage to LDS in row-major [m][64] (padded stride 66)
    #pragma unroll
    for (int nt = 0; nt < 4; ++nt) {
        float bias = sb1[nt * 16 + ln];
        #pragma unroll
        for (int v = 0; v < 8; ++v) {
            int m = v + half * 8;
            h1w[m * 66 + nt * 16 + ln] = leaky(acc1[nt][v] + bias);
        }
    }

    // ---------------- GEMM2: h2 = leaky(h1[16,64] @ W2[64,32] + b2) ----------------
    v8f acc2[2] = {};
    for (int k = 0; k < 64; k += 4) {
        v2f a = *(const v2f*)(h1w + ln * 66 + k + koff);  // A reload in WMMA A-layout
        #pragma unroll
        for (int nt = 0; nt < 2; ++nt) {
            v2f b;
            b.x = sW2[(k + koff)     * 32 + nt * 16 + ln];
            b.y = sW2[(k + koff + 1) * 32 + nt * 16 + ln];
            acc2[nt] = wmma4(a, b, acc2[nt]);
        }
    }
    #pragma unroll
    for (int nt = 0; nt < 2; ++nt) {
        float bias = sb2[nt * 16 + ln];
        #pragma unroll
        for (int v = 0; v < 8; ++v) {
            int m = v + half * 8;
            h2w[m * 34 + nt * 16 + ln] = leaky(acc2[nt][v] + bias);
        }
    }

    // ---------------- GEMM3: mean = h2[16,32] @ W3[32,256] + b3; fused epilogue ----------------
    float part[8] = {0.f, 0.f, 0.f, 0.f, 0.f, 0.f, 0.f, 0.f};  // per-lane noise^2 partials, one per m
    #pragma unroll
    for (int pass = 0; pass < 2; ++pass) {               // 2 x 8 N-tiles keeps VGPRs sane
        v8f acc3[8] = {};
        for (int k = 0; k < 32; k += 4) {
            v2f a = *(const v2f*)(h2w + ln * 34 + k + koff);
            #pragma unroll
            for (int t = 0; t < 8; ++t) {
                int nt = pass * 8 + t;
                v2f b;
                b.x = W3[(k + koff)     * 256 + nt * 16 + ln];
                b.y = W3[(k + koff + 1) * 256 + nt * 16 + ln];
                acc3[t] = wmma4(a, b, acc3[t]);
            }
        }
        #pragma unroll
        for (int t = 0; t < 8; ++t) {
            int nt  = pass * 8 + t;
            int col = nt * 16 + ln;
            float bias = sb3[col];
            #pragma unroll
            for (int v = 0; v < 8; ++v) {
                int row  = r0 + v + half * 8;
                float nz = noise[(size_t)row * 256 + col];
                out_act[(size_t)row * 256 + col] = acc3[t][v] + bias + nz;
                part[v] += nz * nz;                       // z == noise (sigma = 1)
            }
        }
    }

    // log_prob[row] = -0.5 * sum(noise^2) - 256 * 0.5*log(2pi)
    // reduce across the 16 lanes of each half-wave (xor masks 1,2,4,8 stay in-group)
    #pragma unroll
    for (int v = 0; v < 8; ++v) {
        float s = part[v];
        s += __shfl_xor(s, 1, 32);
        s += __shfl_xor(s, 2, 32);
        s += __shfl_xor(s, 4, 32);
        s += __shfl_xor(s, 8, 32);
        if (ln == 0) out_logp[r0 + v + half * 8] = -0.5f * s - LOGP_CONST;
    }
}

extern "C" void kernel_launch(void* const* d_in, const int* in_sizes, int n_in,
                              void* d_out, int out_size, void* d_ws, size_t ws_size,
                              hipStream_t stream) {
    const float* state = (const float*)d_in[0];
    const float* W1    = (const float*)d_in[1];
    const float* b1    = (const float*)d_in[2];
    const float* W2    = (const float*)d_in[3];
    const float* b2    = (const float*)d_in[4];
    const float* W3    = (const float*)d_in[5];
    const float* b3    = (const float*)d_in[6];
    const float* noise = (const float*)d_in[7];

    const int B = in_sizes[0] / 512;                 // 65536
    float* out_act  = (float*)d_out;                 // [B,256]
    float* out_logp = out_act + (size_t)B * 256;     // [B]

    const int rows_per_block = 8 * 16;               // 8 waves x 16-row tiles
    const int blocks = (B + rows_per_block - 1) / rows_per_block;
    policy_mlp_kernel<<<blocks, 256, 0, stream>>>(state, W1, b1, W2, b2, W3, b3,
                                                  noise, out_act, out_logp, B);
}